// MatrixNetAnchors_42949672961320
// MI455X (gfx1250) — compile-verified
//
#include <hip/hip_runtime.h>

typedef unsigned long long u64;
typedef __attribute__((ext_vector_type(2))) float v2f;
typedef __attribute__((ext_vector_type(8))) float v8f;

#define NBLK 32      // stage-A blocks per (map,batch) task
#define NT_A 256     // stage-A block size

// ---------------------------------------------------------------------------
// key packing: descending key order == (score desc, index asc), exactly like
// jax.lax.top_k tie-breaking. pad key 0 sorts below every real entry.
// ---------------------------------------------------------------------------
__device__ __forceinline__ u64 pack_key(float s, unsigned idx) {
  unsigned u = __float_as_uint(s);
  u = (u & 0x80000000u) ? ~u : (u | 0x80000000u);
  return ((u64)u << 32) | (u64)(~idx);
}
__device__ __forceinline__ float unpack_score(u64 k) {
  unsigned u = (unsigned)(k >> 32);
  unsigned b = (u & 0x80000000u) ? (u ^ 0x80000000u) : ~u;
  return __uint_as_float(b);
}
__device__ __forceinline__ unsigned unpack_idx(u64 k) { return ~(unsigned)k; }

// in-LDS bitonic sort, descending; N power of two; multi-element per thread
__device__ __forceinline__ void bitonic_sort_desc(u64* s, int N, int tid, int nthr) {
  for (int k = 2; k <= N; k <<= 1)
    for (int j = k >> 1; j > 0; j >>= 1) {
      __syncthreads();
      for (int i = tid; i < N; i += nthr) {
        int ixj = i ^ j;
        if (ixj > i) {
          u64 a = s[i], b = s[ixj];
          bool sw = ((i & k) == 0) ? (a < b) : (a > b);
          if (sw) { s[i] = b; s[ixj] = a; }
        }
      }
    }
  __syncthreads();
}

// merge a bitonic (desc-then-asc) sequence into descending order
__device__ __forceinline__ void bitonic_merge_desc(u64* s, int N, int tid, int nthr) {
  for (int j = N >> 1; j > 0; j >>= 1) {
    __syncthreads();
    for (int i = tid; i < N; i += nthr) {
      int ixj = i ^ j;
      if (ixj > i) {
        u64 a = s[i], b = s[ixj];
        if (a < b) { s[i] = b; s[ixj] = a; }
      }
    }
  }
  __syncthreads();
}

// ---------------------------------------------------------------------------
// Stage A: fused 3x3 NMS + per-block top-128 over one (map,batch) heat map.
// grid = (NBLK, 2*B); blockIdx.y < B -> tl map, else br map.
// ---------------------------------------------------------------------------
__global__ __launch_bounds__(NT_A)
void nms_topk_part(const float* __restrict__ tl, const float* __restrict__ br,
                   int cat, int h, int w, u64* __restrict__ cand) {
  const int task = blockIdx.y;                 // 0..15  (map*B + b)
  const int bi   = (task < 8) ? task : task - 8;
  const float* heat = ((task < 8) ? tl : br) + (size_t)bi * cat * h * w;
  const int hw = h * w;
  const int n  = cat * hw;
  const int tid = threadIdx.x;

  __shared__ u64 best[128];
  __shared__ u64 cbuf[NT_A];
  __shared__ u64 mrg[NT_A];
  __shared__ int flag;

  if (tid < 128) best[tid] = 0ull;
  __syncthreads();

  for (int base = blockIdx.x * NT_A; base < n; base += NBLK * NT_A) {
    const int idx = base + tid;
    u64 key = 0ull;
    if (idx < n) {
      const int c   = idx / hw;
      const int rem = idx - c * hw;
      const int y = rem / w;
      const int x = rem - y * w;
      const float* hc = heat + (size_t)c * hw;
      const float v = hc[rem];
      float m = v;
      const int y0 = y > 0 ? y - 1 : 0, y1 = y < h - 1 ? y + 1 : y;
      const int x0 = x > 0 ? x - 1 : 0, x1 = x < w - 1 ? x + 1 : x;
      for (int yy = y0; yy <= y1; ++yy)
        for (int xx = x0; xx <= x1; ++xx)
          m = fmaxf(m, hc[yy * w + xx]);
      const float s = (v >= m) ? v : 0.0f;     // heat * (hmax == heat)
      key = pack_key(s, (unsigned)idx);
      if (idx + NBLK * NT_A < n) __builtin_prefetch(heat + idx + NBLK * NT_A, 0, 0);
    }
    const u64 thr = best[127];
    if (tid == 0) flag = 0;
    __syncthreads();
    if (key > thr) flag = 1;
    __syncthreads();
    if (flag) {                                // uniform (shared) condition
      cbuf[tid] = key;
      bitonic_sort_desc(cbuf, NT_A, tid, NT_A);
      mrg[tid] = (tid < 128) ? best[tid] : cbuf[255 - tid];   // desc + asc = bitonic
      bitonic_merge_desc(mrg, 256, tid, NT_A);
      if (tid < 128) best[tid] = mrg[tid];
    }
    __syncthreads();
  }
  if (tid < 128)
    cand[((size_t)task * NBLK + blockIdx.x) * 128 + tid] = best[tid];
}

// ---------------------------------------------------------------------------
// Stage B: merge NBLK*128 = 4096 candidates -> sorted top-128 per task
// ---------------------------------------------------------------------------
__global__ __launch_bounds__(1024)
void topk_merge(const u64* __restrict__ cand, u64* __restrict__ topk) {
  __shared__ u64 s[NBLK * 128];
  const int t = blockIdx.x, tid = threadIdx.x;
  for (int i = tid; i < NBLK * 128; i += 1024)
    s[i] = cand[(size_t)t * (NBLK * 128) + i];
  bitonic_sort_desc(s, NBLK * 128, tid, 1024);
  if (tid < 128) topk[(size_t)t * 128 + tid] = s[tid];
}

// ---------------------------------------------------------------------------
// pairwise scoring + top-1000 per (level, batch)
// ---------------------------------------------------------------------------
__device__ __forceinline__ void emit_pair(
    int i, int j, float sum, int hw_unused,
    const float* tlx, const float* tly, const float* tlcx, const float* tlcy,
    const float* tls, const int* tlcl,
    const float* brx, const float* bry, const float* brcx, const float* brcy,
    const float* brs, const int* brcl,
    float wmin08, float wmax13, float hmin08, float hmax13,
    u64* sortbuf) {
  const float wd = brx[j] - tlx[i];
  const float ht = bry[j] - tly[i];
  const float score = 0.5f * sum;
  bool bad = (tlcl[i] != brcl[j]);
  bad = bad | (wd < wmin08) | (wd > wmax13) | (ht < hmin08) | (ht > hmax13);
  const float dx = fabsf(1.0f - 8.0f * (tlcx[i] + brcx[j]) / wd);   // W_LAST = 8
  const float dy = fabsf(1.0f - 8.0f * (tlcy[i] + brcy[j]) / ht);   // H_LAST = 8
  const float dd = fabsf(brcx[j] - tlcx[i]) + fabsf(brcy[j] - tlcy[i]);
  bad = bad | (dx > 0.2f) | (dy > 0.2f) | (dd > 0.25f) | (wd < 0.0f) | (ht < 0.0f);
  const float s = bad ? -1.0f : score;
  const int p = i * 100 + j;
  sortbuf[p] = pack_key(s, (unsigned)p);
}

__global__ __launch_bounds__(1024)
void pairs_dets(const float* __restrict__ tl_regr, const float* __restrict__ br_regr,
                const float* __restrict__ tl_ctr,  const float* __restrict__ br_ctr,
                const u64* __restrict__ topk,      // [2][B][128] for this level
                float* __restrict__ dets,          // [B][5000][8]
                int lvl, int h, int w,
                float wmin08, float wmax13, float hmin08, float hmax13,
                float wscale, float hscale) {
  extern __shared__ u64 sortbuf[];                 // 16384 keys (128 KB)
  __shared__ float tlx[112], tly[112], tlcx[112], tlcy[112], tls[112];
  __shared__ float brx[112], bry[112], brcx[112], brcy[112], brs[112];
  __shared__ int   tlcl[112], brcl[112];

  const int b = blockIdx.x, tid = threadIdx.x;
  const int hw = h * w;

  // unpack top-100 of each map into LDS (pad to 112 with zeros)
  const int m  = tid >> 9;       // 0 = tl, 1 = br
  const int k2 = tid & 511;
  if (k2 < 112) {
    float sx = 0.f, sy = 0.f, cx = 0.f, cy = 0.f, sc = 0.f; int cl = 0;
    if (k2 < 100) {
      const u64 key = topk[((size_t)m * 8 + b) * 128 + k2];
      sc = unpack_score(key);
      const unsigned idx = unpack_idx(key);
      const int c   = (int)(idx / (unsigned)hw);
      const int rem = (int)(idx - (unsigned)c * (unsigned)hw);
      const int y = rem / w;
      const int x = rem - y * w;
      const float* regr = (m == 0 ? tl_regr : br_regr) + (size_t)b * 2 * hw;
      const float* ctr  = (m == 0 ? tl_ctr  : br_ctr ) + (size_t)b * 2 * hw;
      sx = (float)x + regr[rem];
      sy = (float)y + regr[hw + rem];
      cx = ctr[rem];
      cy = ctr[hw + rem];
      cl = c;
    }
    if (m == 0) { tlx[k2]=sx; tly[k2]=sy; tlcx[k2]=cx; tlcy[k2]=cy; tls[k2]=sc; tlcl[k2]=cl; }
    else        { brx[k2]=sx; bry[k2]=sy; brcx[k2]=cx; brcy[k2]=cy; brs[k2]=sc; brcl[k2]=cl; }
  }
  for (int i = tid; i < 16384; i += 1024) sortbuf[i] = 0ull;   // pad keys
  __syncthreads();

#if defined(__gfx1250__) && __has_builtin(__builtin_amdgcn_wmma_f32_16x16x4_f32)
  // base score tls[i]+brs[j] via V_WMMA_F32_16X16X4_F32 (exact in f32:
  // tls*1 + 1*brs + 0*0 + 0*0). 7x7 tiles of 16x16 cover the 100x100 grid.
  {
    const int wv = tid >> 5, lane = tid & 31;
    const bool lo = lane < 16;
    for (int t = wv; t < 49; t += 32) {            // wave-uniform trip count
      const int ti = t / 7, tj = t - ti * 7;
      v2f a, bm;
      a.x  = lo ? tls[ti * 16 + lane] : 0.0f;      // A[m][0] / A[m][2]
      a.y  = lo ? 1.0f                : 0.0f;      // A[m][1] / A[m][3]
      bm.x = lo ? 1.0f                : 0.0f;      // B[0][n] / B[2][n]
      bm.y = lo ? brs[tj * 16 + lane] : 0.0f;      // B[1][n] / B[3][n]
      v8f c = {0.f,0.f,0.f,0.f,0.f,0.f,0.f,0.f};
      c = __builtin_amdgcn_wmma_f32_16x16x4_f32(false, a, false, bm,
                                                (short)0, c, false, false);
      const int j     = tj * 16 + (lane & 15);
      const int ibase = ti * 16 + ((lane & 16) ? 8 : 0);
#pragma unroll
      for (int r = 0; r < 8; ++r) {
        const int i = ibase + r;
        if (i < 100 && j < 100)
          emit_pair(i, j, c[r], hw, tlx, tly, tlcx, tlcy, tls, tlcl,
                    brx, bry, brcx, brcy, brs, brcl,
                    wmin08, wmax13, hmin08, hmax13, sortbuf);
      }
    }
  }
#else
  for (int p = tid; p < 10000; p += 1024) {
    const int i = p / 100, j = p - i * 100;
    emit_pair(i, j, tls[i] + brs[j], hw, tlx, tly, tlcx, tlcy, tls, tlcl,
              brx, bry, brcx, brcy, brs, brcl,
              wmin08, wmax13, hmin08, hmax13, sortbuf);
  }
#endif
  __syncthreads();
  bitonic_sort_desc(sortbuf, 16384, tid, 1024);

  if (tid < 1000) {
    const u64 key = sortbuf[tid];
    const unsigned p = unpack_idx(key);
    const float s = unpack_score(key);
    const int i = (int)(p / 100u), j = (int)(p - (unsigned)i * 100u);
    float* row = dets + ((size_t)b * 5000 + (size_t)lvl * 1000 + tid) * 8;
    row[0] = tlx[i] * wscale;
    row[1] = tly[i] * hscale;
    row[2] = brx[j] * wscale;
    row[3] = bry[j] * hscale;
    row[4] = s;
    row[5] = tls[i];
    row[6] = brs[j];
    row[7] = (float)tlcl[i];
  }
}

// ---------------------------------------------------------------------------
// final per-batch descending sort of 5000 detections by score (col 4)
// ---------------------------------------------------------------------------
__global__ __launch_bounds__(1024)
void final_sort(const float* __restrict__ dets, float* __restrict__ out) {
  extern __shared__ u64 s[];                       // 8192 keys (64 KB)
  const int b = blockIdx.x, tid = threadIdx.x;
  const float* db = dets + (size_t)b * 5000 * 8;
  for (int i = tid; i < 8192; i += 1024)
    s[i] = (i < 5000) ? pack_key(db[(size_t)i * 8 + 4], (unsigned)i) : 0ull;
  bitonic_sort_desc(s, 8192, tid, 1024);
  for (int r = tid; r < 5000; r += 1024) {
    const unsigned pos = unpack_idx(s[r]);
    const float* src = db + (size_t)pos * 8;
    float* o = out + ((size_t)b * 5000 + r) * 8;
#pragma unroll
    for (int c = 0; c < 8; ++c) o[c] = src[c];
  }
}

// ---------------------------------------------------------------------------
extern "C" void kernel_launch(void* const* d_in, const int* in_sizes, int n_in,
                              void* d_out, int out_size, void* d_ws, size_t ws_size,
                              hipStream_t stream) {
  (void)in_sizes; (void)n_in; (void)out_size; (void)ws_size;
  const int B = 8, CAT = 80;
  static const int HH[5] = {128, 64, 32, 16, 8};
  // (hmin, hmax, wmin, wmax) per level; thresholds computed in f64 then cast,
  // matching JAX's python-scalar * f32 promotion.
  static const double RR[5][4] = {{10, 128, 10, 128}, {8, 64, 8, 64},
                                  {6, 32, 6, 32},     {4, 16, 4, 16},
                                  {2, 8, 2, 8}};
  char* ws = (char*)d_ws;
  u64*   topkKeys = (u64*)ws;                          // 5*2*8*128*8  = 80 KB
  float* dets     = (float*)(ws + 81920);              // 8*5000*8*4   = 1.25 MB
  u64*   cand     = (u64*)(ws + 81920 + 1280000);      // 16*32*128*8  = 512 KB

  for (int l = 0; l < 5; ++l) {
    const float* tlh = (const float*)d_in[l * 6 + 0];
    const float* brh = (const float*)d_in[l * 6 + 1];
    const float* tlr = (const float*)d_in[l * 6 + 2];
    const float* brr = (const float*)d_in[l * 6 + 3];
    const float* tlc = (const float*)d_in[l * 6 + 4];
    const float* brc = (const float*)d_in[l * 6 + 5];
    const int h = HH[l], w = HH[l];

    nms_topk_part<<<dim3(NBLK, 2 * B), NT_A, 0, stream>>>(tlh, brh, CAT, h, w, cand);
    topk_merge<<<dim3(2 * B), 1024, 0, stream>>>(cand, topkKeys + (size_t)l * 2 * B * 128);

    const double hmin = RR[l][0], hmax = RR[l][1], wmin = RR[l][2], wmax = RR[l][3];
    pairs_dets<<<dim3(B), 1024, 16384 * sizeof(u64), stream>>>(
        tlr, brr, tlc, brc, topkKeys + (size_t)l * 2 * B * 128, dets, l, h, w,
        (float)(0.8 * wmin), (float)(1.3 * wmax),
        (float)(0.8 * hmin), (float)(1.3 * hmax),
        (float)(128.0 / w), (float)(128.0 / h));
  }
  final_sort<<<dim3(B), 1024, 8192 * sizeof(u64), stream>>>(dets, (float*)d_out);
}